// TopoAE_42563125903700
// MI455X (gfx1250) — compile-verified
//
#include <hip/hip_runtime.h>
#include <hip/hip_bf16.h>
#include <cstddef>

// ---------------------------------------------------------------------------
// TopoAE loss on MI455X (gfx1250): fp32 WMMA (16x16x4) for all GEMM/Gram work,
// fused epilogues (relu/bias, rec-loss, cdist), Prim's MST in LDS.
// ---------------------------------------------------------------------------

typedef __attribute__((ext_vector_type(2))) float v2f;
typedef __attribute__((ext_vector_type(8))) float v8f;

#define BB 2048
#define DD 4096
#define HH 1024
#define LL 64

// -------------------------- small helper kernels ---------------------------

__global__ void init_kernel(unsigned* xmaxBits) {
    if (threadIdx.x == 0) *xmaxBits = 0u;
}

__global__ void sqnorm_kernel(const float* __restrict__ X, int K, float* __restrict__ out) {
    __shared__ float red[256];
    const int row = blockIdx.x;
    float s = 0.0f;
    for (int c = threadIdx.x; c < K; c += blockDim.x) {
        float v = X[(size_t)row * K + c];
        s += v * v;
    }
    red[threadIdx.x] = s;
    __syncthreads();
    for (int st = 128; st > 0; st >>= 1) {
        if ((int)threadIdx.x < st) red[threadIdx.x] += red[threadIdx.x + st];
        __syncthreads();
    }
    if (threadIdx.x == 0) out[row] = red[0];
}

__global__ void finalize_kernel(const float* __restrict__ recP, const float* __restrict__ topoP,
                                float* __restrict__ out) {
    float rec = 0.0f;
    for (int i = 0; i < 1024; ++i) rec += recP[i];
    float topo = topoP[0] + topoP[1];
    out[0] = rec / ((float)BB * (float)DD) + 1.0f * (topo / (float)BB);
}

// ------------------------------ WMMA GEMM ----------------------------------
// C = A(MxK) @ B(KxN). Block tile 128x64, 8 waves (4M x 2N), wave tile 32x32
// as 2x2 fragments of V_WMMA_F32_16X16X4_F32. B staged transposed in LDS so
// both A and B fragments are contiguous 8B (ds_load_b64) reads, bank-conflict
// free (row stride 68 dwords -> rows land on banks spaced 4 apart).
// Register-buffered pipeline overlaps next tile's global loads with WMMA.
//
// EPI: 0 = relu(acc+bias) store, 1 = (acc+bias) store,
//      2 = rec-loss: sum((acc+bias - aux0[row,col])^2) -> partial[block]
//      3 = dist: d = sqrt(max(aux0[row]+aux0[col]-2*acc,0)) store + atomicMax
// BT:  B element [k][n] is read as X[n][k] (Gram / A@A^T mode).

template <int EPI, bool BT>
__global__ __launch_bounds__(256) void gemm_wmma(
    const float* __restrict__ A, const float* __restrict__ B,
    const float* __restrict__ bias, float* __restrict__ out,
    const float* __restrict__ aux0, float* __restrict__ partials,
    unsigned* accU, int M, int N, int K) {

    __shared__ float As[128][68];   // A  tile, [m][k], stride 68 (16B-aligned rows)
    __shared__ float BsT[64][68];   // B^T tile, [n][k], stride 68
    __shared__ float red[256];

    const int tid   = threadIdx.x;
    const int lane  = tid & 31;
    const int wave  = tid >> 5;
    const int wm    = wave >> 1;      // 0..3
    const int wn    = wave & 1;       // 0..1
    const int lhalf = lane >> 4;      // 0,1
    const int lmod  = lane & 15;
    const int bM    = blockIdx.x * 128;
    const int bN    = blockIdx.y * 64;

    v8f acc[2][2] = {};
    float4 aReg[8];
    float4 bReg[4];

    auto loadTile = [&](int k0) {
        #pragma unroll
        for (int it = 0; it < 8; ++it) {
            int idx = tid + it * 256;
            aReg[it] = *(const float4*)&A[(size_t)(bM + (idx >> 4)) * K + k0 + ((idx & 15) << 2)];
        }
        #pragma unroll
        for (int it = 0; it < 4; ++it) {
            int idx = tid + it * 256;
            if (BT) {
                // row = n, cols = k (x rows are k-contiguous)
                bReg[it] = *(const float4*)&B[(size_t)(bN + (idx >> 4)) * K + k0 + ((idx & 15) << 2)];
            } else {
                // row = k, cols = n (W rows are n-contiguous)
                bReg[it] = *(const float4*)&B[(size_t)(k0 + (idx >> 4)) * N + bN + ((idx & 15) << 2)];
            }
        }
    };
    auto storeTile = [&]() {
        #pragma unroll
        for (int it = 0; it < 8; ++it) {
            int idx = tid + it * 256;
            *(float4*)&As[idx >> 4][(idx & 15) << 2] = aReg[it];
        }
        #pragma unroll
        for (int it = 0; it < 4; ++it) {
            int idx = tid + it * 256;
            int r = idx >> 4, c4 = (idx & 15) << 2;
            if (BT) {
                *(float4*)&BsT[r][c4] = bReg[it];          // already [n][k]
            } else {
                BsT[c4 + 0][r] = bReg[it].x;               // transpose scatter
                BsT[c4 + 1][r] = bReg[it].y;
                BsT[c4 + 2][r] = bReg[it].z;
                BsT[c4 + 3][r] = bReg[it].w;
            }
        }
    };

    loadTile(0);
    for (int k0 = 0; k0 < K; k0 += 64) {
        storeTile();
        __syncthreads();
        if (k0 + 64 < K) loadTile(k0 + 64);   // overlaps with WMMA below

        #pragma unroll
        for (int kk = 0; kk < 64; kk += 4) {
            // A frag (16x4 f32): lane -> M=lmod, K = 2*lhalf + v
            v2f a0 = *(const v2f*)&As[wm * 32 + lmod][kk + 2 * lhalf];
            v2f a1 = *(const v2f*)&As[wm * 32 + 16 + lmod][kk + 2 * lhalf];
            // B frag (4x16 f32): lane -> N=lmod, K = 2*lhalf + v (contiguous in BsT)
            v2f b0 = *(const v2f*)&BsT[wn * 32 + lmod][kk + 2 * lhalf];
            v2f b1 = *(const v2f*)&BsT[wn * 32 + 16 + lmod][kk + 2 * lhalf];

            acc[0][0] = __builtin_amdgcn_wmma_f32_16x16x4_f32(false, a0, false, b0, (short)0, acc[0][0], false, false);
            acc[0][1] = __builtin_amdgcn_wmma_f32_16x16x4_f32(false, a0, false, b1, (short)0, acc[0][1], false, false);
            acc[1][0] = __builtin_amdgcn_wmma_f32_16x16x4_f32(false, a1, false, b0, (short)0, acc[1][0], false, false);
            acc[1][1] = __builtin_amdgcn_wmma_f32_16x16x4_f32(false, a1, false, b1, (short)0, acc[1][1], false, false);
        }
        __syncthreads();
    }

    // ------------------------------ epilogue -------------------------------
    const int rowBase = bM + wm * 32;
    const int colBase = bN + wn * 32;
    float localAcc = 0.0f;
    float localMax = 0.0f;

    #pragma unroll
    for (int mi = 0; mi < 2; ++mi) {
        #pragma unroll
        for (int ni = 0; ni < 2; ++ni) {
            #pragma unroll
            for (int v = 0; v < 8; ++v) {
                // C/D layout: row = v + 8*lhalf, col = lmod
                int row = rowBase + mi * 16 + v + 8 * lhalf;
                int col = colBase + ni * 16 + lmod;
                float val = acc[mi][ni][v];
                if (EPI == 0) {
                    val += bias[col];
                    out[(size_t)row * N + col] = fmaxf(val, 0.0f);
                } else if (EPI == 1) {
                    val += bias[col];
                    out[(size_t)row * N + col] = val;
                } else if (EPI == 2) {
                    val += bias[col];
                    float r = val - aux0[(size_t)row * N + col];
                    localAcc += r * r;
                } else {
                    float d2 = aux0[row] + aux0[col] - 2.0f * val;
                    float d  = d2 > 0.0f ? sqrtf(d2) : 0.0f;
                    out[(size_t)row * N + col] = d;
                    localMax = fmaxf(localMax, d);
                }
            }
        }
    }

    if (EPI == 2) {
        red[tid] = localAcc;
        __syncthreads();
        for (int s = 128; s > 0; s >>= 1) {
            if (tid < s) red[tid] += red[tid + s];
            __syncthreads();
        }
        if (tid == 0) partials[blockIdx.y * gridDim.x + blockIdx.x] = red[0];
    }
    if (EPI == 3) {
        red[tid] = localMax;
        __syncthreads();
        for (int s = 128; s > 0; s >>= 1) {
            if (tid < s) red[tid] = fmaxf(red[tid], red[tid + s]);
            __syncthreads();
        }
        if (tid == 0 && accU != nullptr) atomicMax(accU, __float_as_uint(red[0]));
    }
}

// --------------------------------- MST -------------------------------------
// Prim's over n=2048 nodes, one block per tree. block 0: x-tree, block 1:
// z-tree. Signature accumulated directly (scaling applied here only; MST
// topology is invariant under positive scaling of d).

__global__ __launch_bounds__(1024) void mst_topo_kernel(
    const float* __restrict__ dx, const float* __restrict__ dz,
    const unsigned* __restrict__ xmaxBits, const float* __restrict__ latentNorm,
    float* __restrict__ topoP) {

    const int n = BB;
    __shared__ float minDist[BB];
    __shared__ int   parent[BB];
    __shared__ unsigned inTree[BB / 32];
    __shared__ float redV[1024];
    __shared__ int   redI[1024];
    __shared__ float sigSh;

    const float xmax = __uint_as_float(*xmaxBits);
    const float ln   = latentNorm[0];
    const float* dP;
    const float* dO;
    float rsp, rso;
    if (blockIdx.x == 0) { dP = dx; dO = dz; rsp = 1.0f / xmax; rso = 1.0f / ln; }
    else                 { dP = dz; dO = dx; rsp = 1.0f / ln;   rso = 1.0f / xmax; }

    const int t = threadIdx.x;
    for (int k = t; k < n; k += 1024) { minDist[k] = dP[k]; parent[k] = 0; }
    for (int w = t; w < n / 32; w += 1024) inTree[w] = (w == 0) ? 1u : 0u;
    if (t == 0) sigSh = 0.0f;
    __syncthreads();

    for (int step = 0; step < n - 1; ++step) {
        // argmin over not-in-tree (first-index tie-break, like jnp.argmin)
        float bv = __builtin_inff();
        int   bi = n;
        #pragma unroll
        for (int q = 0; q < 2; ++q) {
            int k = t + q * 1024;
            bool it = (inTree[k >> 5] >> (k & 31)) & 1u;
            float v = it ? __builtin_inff() : minDist[k];
            if (v < bv || (v == bv && k < bi)) { bv = v; bi = k; }
        }
        redV[t] = bv;
        redI[t] = bi;
        __syncthreads();
        for (int s = 512; s >= 1; s >>= 1) {
            if (t < s) {
                float ov = redV[t + s];
                int   oi = redI[t + s];
                if (ov < redV[t] || (ov == redV[t] && oi < redI[t])) {
                    redV[t] = ov;
                    redI[t] = oi;
                }
            }
            __syncthreads();
        }
        const int j = redI[0];
        if (t == 0) {
            int i = parent[j];
            inTree[j >> 5] |= (1u << (j & 31));
            float dpij = dP[(size_t)i * n + j] * rsp;
            float doij = dO[(size_t)i * n + j] * rso;
            float dlt  = dpij - doij;
            sigSh += dlt * dlt;
        }
        __syncthreads();
        const float* rowj = dP + (size_t)j * n;
        #pragma unroll
        for (int q = 0; q < 2; ++q) {
            int k = t + q * 1024;
            bool it = (inTree[k >> 5] >> (k & 31)) & 1u;
            if (!it) {
                float v = rowj[k];
                if (v < minDist[k]) { minDist[k] = v; parent[k] = j; }
            }
        }
        __syncthreads();
    }
    if (t == 0) topoP[blockIdx.x] = sigSh;
}

// ------------------------------- launcher ----------------------------------

extern "C" void kernel_launch(void* const* d_in, const int* in_sizes, int n_in,
                              void* d_out, int out_size, void* d_ws, size_t ws_size,
                              hipStream_t stream) {
    const float* x   = (const float*)d_in[0];
    const float* W1  = (const float*)d_in[1];
    const float* b1  = (const float*)d_in[2];
    const float* W2  = (const float*)d_in[3];
    const float* b2  = (const float*)d_in[4];
    const float* W3  = (const float*)d_in[5];
    const float* b3  = (const float*)d_in[6];
    const float* W4  = (const float*)d_in[7];
    const float* b4  = (const float*)d_in[8];
    const float* lnorm = (const float*)d_in[9];
    float* out = (float*)d_out;

    char* w = (char*)d_ws;
    float*    recP     = (float*)w;                 // 1024 floats (one per recloss block)
    float*    topoP    = recP + 1024;               // 2 floats
    unsigned* xmaxBits = (unsigned*)(topoP + 2);    // 1 uint
    float*    sqx      = (float*)(w + 8192);        // 2048
    float*    sqz      = sqx + BB;                  // 2048
    float*    z        = sqz + BB;                  // 2048*64
    float*    h        = z + (size_t)BB * LL;       // 2048*1024 (h1, then h2)
    float*    xd       = h + (size_t)BB * HH;       // 2048*2048
    float*    zd       = xd + (size_t)BB * BB;      // 2048*2048

    init_kernel<<<1, 32, 0, stream>>>(xmaxBits);

    // squared row norms of x
    sqnorm_kernel<<<BB, 256, 0, stream>>>(x, DD, sqx);

    // h1 = relu(x @ W1 + b1)          M=2048 N=1024 K=4096
    gemm_wmma<0, false><<<dim3(BB / 128, HH / 64), 256, 0, stream>>>(
        x, W1, b1, h, nullptr, nullptr, nullptr, BB, HH, DD);

    // z = h1 @ W2 + b2                M=2048 N=64 K=1024
    gemm_wmma<1, false><<<dim3(BB / 128, LL / 64), 256, 0, stream>>>(
        h, W2, b2, z, nullptr, nullptr, nullptr, BB, LL, HH);

    sqnorm_kernel<<<BB, 256, 0, stream>>>(z, LL, sqz);

    // h2 = relu(z @ W3 + b3)          M=2048 N=1024 K=64 (overwrites h1)
    gemm_wmma<0, false><<<dim3(BB / 128, HH / 64), 256, 0, stream>>>(
        z, W3, b3, h, nullptr, nullptr, nullptr, BB, HH, LL);

    // rec partials from (h2 @ W4 + b4 - x)^2    M=2048 N=4096 K=1024
    gemm_wmma<2, false><<<dim3(BB / 128, DD / 64), 256, 0, stream>>>(
        h, W4, b4, nullptr, x, recP, nullptr, BB, DD, HH);

    // x_dist (unnormalized) + max     M=N=2048 K=4096 (Gram of x)
    gemm_wmma<3, true><<<dim3(BB / 128, BB / 64), 256, 0, stream>>>(
        x, x, nullptr, xd, sqx, nullptr, xmaxBits, BB, BB, DD);

    // z_dist (unnormalized)           M=N=2048 K=64 (Gram of z)
    gemm_wmma<3, true><<<dim3(BB / 128, BB / 64), 256, 0, stream>>>(
        z, z, nullptr, zd, sqz, nullptr, nullptr, BB, BB, LL);

    // two MSTs + topo signature
    mst_topo_kernel<<<2, 1024, 0, stream>>>(xd, zd, xmaxBits, lnorm, topoP);

    finalize_kernel<<<1, 1, 0, stream>>>(recP, topoP, out);
}